// TripletLoss_77627238908257
// MI455X (gfx1250) — compile-verified
//
#include <hip/hip_runtime.h>
#include <math.h>

typedef __attribute__((ext_vector_type(2))) float v2f;
typedef __attribute__((ext_vector_type(8))) float v8f;

#define B_ 32
#define N_ 2048
#define D_ 128
#define MARGIN_ 0.1f
#define EPS_ 1e-12f

// ---------------------------------------------------------------------------
// Kernel 1: dist[b,n] = || a_hat(b) - f_hat(b,n) ||  via WMMA f32 16x16x4.
// Grid: (N/128, B), block 256 (8 waves; each wave owns a 16-row tile).
// A operand = 16x4 feature chunk, B operand = anchor broadcast to 16 cols.
// ---------------------------------------------------------------------------
__global__ __launch_bounds__(256)
void triplet_dist_kernel(const float* __restrict__ anchors,
                         const float* __restrict__ feats,
                         float* __restrict__ dist) {
    __shared__ float sAhat[D_];
    __shared__ float red[128];
    __shared__ float sAnc2;

    const int b   = blockIdx.y;
    const int tid = threadIdx.x;

    // --- normalize anchor into LDS (deterministic tree reduction) ---
    if (tid < D_) {
        float v = anchors[b * D_ + tid];
        sAhat[tid] = v;
        red[tid]   = v * v;
    }
    __syncthreads();
    for (int s = 64; s > 0; s >>= 1) {
        if (tid < s) red[tid] += red[tid + s];
        __syncthreads();
    }
    const float nrm2a = red[0];
    const float invA  = 1.0f / (sqrtf(nrm2a) + EPS_);
    if (tid == 0) sAnc2 = nrm2a * invA * invA;   // |a_hat|^2
    __syncthreads();
    if (tid < D_) sAhat[tid] *= invA;
    __syncthreads();

    // --- per-wave 16-row tile ---
    const int wave = tid >> 5;
    const int lane = tid & 31;
    const int g    = lane >> 4;      // K-half selector per ISA A layout
    const int m    = lane & 15;      // row within tile
    const int n0   = blockIdx.x * 128 + wave * 16;

    const float* rowp = feats + ((size_t)b * N_ + (size_t)(n0 + m)) * D_;

    v8f  acc  = {0.f, 0.f, 0.f, 0.f, 0.f, 0.f, 0.f, 0.f};
    float nrm2 = 0.f;

#pragma unroll 4
    for (int k0 = 0; k0 < D_; k0 += 4) {
        const int kb = k0 + 2 * g;                 // lanes 0-15: K={0,1}; 16-31: K={2,3}
        v2f av = *(const v2f*)(rowp + kb);         // global_load_b64
        v2f bv;
        bv.x = sAhat[kb];
        bv.y = sAhat[kb + 1];
        acc = __builtin_amdgcn_wmma_f32_16x16x4_f32(
                  false, av, false, bv, (short)0, acc, false, false);
        nrm2 = fmaf(av.x, av.x, nrm2);
        nrm2 = fmaf(av.y, av.y, nrm2);
    }

    // merge the two K-halves of the feature norm (wave32 xor-16 shuffle)
    float nrm2f = nrm2 + __shfl_xor(nrm2, 16, 32);

    // C/D layout: VGPR r, lane l -> row M = r + 8*(l>>4), col = l&15 (all cols equal).
    const int r = lane & 7;
    const int M = (g << 3) + r;
    float nr = __shfl(nrm2f, M, 32);               // row-M feature norm^2

    float dot = acc[0];
#pragma unroll
    for (int q = 1; q < 8; ++q) dot = (r == q) ? acc[q] : dot;

    const float sf   = sqrtf(nr) + EPS_;
    const float invf = 1.0f / sf;
    float sq = sAnc2 + nr * invf * invf - 2.0f * dot * invf;
    sq = fmaxf(sq, 1e-12f);
    // lanes l and l+8 in the same half write identical data to the same row: benign
    dist[(size_t)b * N_ + (size_t)(n0 + M)] = sqrtf(sq);
}

// ---------------------------------------------------------------------------
// Kernel 2: partial[b,s] = sum over positives i in slice s, negatives j of
//           relu(d_i - d_j + margin).  Grid (16, B), block 256.
// ---------------------------------------------------------------------------
__global__ __launch_bounds__(256)
void triplet_pair_kernel(const float* __restrict__ dist,
                         const int* __restrict__ scores,
                         float* __restrict__ partial,
                         int* __restrict__ cntpos) {
    __shared__ float sdn[N_];   // dist if negative, +BIG otherwise
    __shared__ float sdr[N_];   // raw dist
    __shared__ int   ssc[N_];
    __shared__ float red[256];

    const int b   = blockIdx.y;
    const int tid = threadIdx.x;

    for (int idx = tid; idx < N_; idx += 256) {
        float d  = dist[(size_t)b * N_ + idx];
        int   sc = scores[(size_t)b * N_ + idx];
        sdr[idx] = d;
        ssc[idx] = sc;
        sdn[idx] = (sc == 0) ? d : 3.0e38f;        // relu(th - BIG) == 0
    }
    __syncthreads();

    const int i    = blockIdx.x * 128 + (tid & 127);
    const int half = tid >> 7;
    float acc = 0.f;
    if (ssc[i] > 0) {
        const float th = sdr[i] + MARGIN_;
#pragma unroll 8
        for (int j = half; j < N_; j += 2) {
            acc += fmaxf(0.f, th - sdn[j]);
        }
    }
    red[tid] = acc;
    __syncthreads();
    for (int s = 128; s > 0; s >>= 1) {
        if (tid < s) red[tid] += red[tid + s];
        __syncthreads();
    }
    if (tid == 0) partial[b * 16 + blockIdx.x] = red[0];

    // one block per row counts positives (scores are {0,1} -> n_neg = N - n_pos)
    if (blockIdx.x == 0) {
        __syncthreads();
        int c = 0;
        for (int idx = tid; idx < N_; idx += 256) c += (ssc[idx] > 0) ? 1 : 0;
        red[tid] = (float)c;                       // exact: counts <= 2048
        __syncthreads();
        for (int s = 128; s > 0; s >>= 1) {
            if (tid < s) red[tid] += red[tid + s];
            __syncthreads();
        }
        if (tid == 0) cntpos[b] = (int)red[0];
    }
}

// ---------------------------------------------------------------------------
// Kernel 3: per-row normalization + final deterministic sum.  1 block, 32 lanes.
// ---------------------------------------------------------------------------
__global__ void triplet_final_kernel(const float* __restrict__ partial,
                                     const int* __restrict__ cntpos,
                                     float* __restrict__ out) {
    const int lane = threadIdx.x;                  // 0..31 == row b
    float sum = 0.f;
    for (int s = 0; s < 16; ++s) sum += partial[lane * 16 + s];
    const int np = cntpos[lane];
    const int nn = N_ - np;
    const float denom = fmaxf((float)(np * nn), 1.0f);
    float row = (np > 0) ? (sum / denom) : MARGIN_;
    for (int off = 16; off > 0; off >>= 1)
        row += __shfl_down(row, off, 32);
    if (lane == 0) out[0] = row;
}

extern "C" void kernel_launch(void* const* d_in, const int* in_sizes, int n_in,
                              void* d_out, int out_size, void* d_ws, size_t ws_size,
                              hipStream_t stream) {
    const float* anchors = (const float*)d_in[0];   // [B,D]
    const float* feats   = (const float*)d_in[1];   // [B,N,D]
    const int*   scores  = (const int*)d_in[2];     // [B,N]
    float* out = (float*)d_out;

    float* dist    = (float*)d_ws;                  // B*N floats
    float* partial = dist + (size_t)B_ * N_;        // B*16 floats
    int*   cnt     = (int*)(partial + B_ * 16);     // B ints

    triplet_dist_kernel<<<dim3(N_ / 128, B_), 256, 0, stream>>>(anchors, feats, dist);
    triplet_pair_kernel<<<dim3(16, B_), 256, 0, stream>>>(dist, scores, partial, cnt);
    triplet_final_kernel<<<1, 32, 0, stream>>>(partial, cnt, out);
}